// GNN_45775761440951
// MI455X (gfx1250) — compile-verified
//
#include <hip/hip_runtime.h>

#define NN      100000
#define EE      1600000
#define EPRIME  1700000     // EE + NN (with self loops)
#define FNODE   128
#define FEDGE   32
#define HID     64
#define ACTD    32
#define REPEAT  100         // N / NG
#define NIDX    1000
#define NEG     0.2f
#define NTILES  (EPRIME / 16)   // 106250, exact

typedef float v2f __attribute__((ext_vector_type(2)));
typedef float v8f __attribute__((ext_vector_type(8)));

__device__ __forceinline__ v8f wmma4(v2f a, v2f b, v8f c) {
  // V_WMMA_F32_16X16X4_F32 : D = A(16x4) * B(4x16) + C(16x16), all fp32
  return __builtin_amdgcn_wmma_f32_16x16x4_f32(
      /*neg_a=*/false, a, /*neg_b=*/false, b,
      /*c_mod=*/(short)0, c, /*reuse_a=*/false, /*reuse_b=*/false);
}

__device__ __forceinline__ void atomicMaxF(float* addr, float v) {
  if (v >= 0.0f) atomicMax((int*)addr, __float_as_int(v));
  else           atomicMin((unsigned int*)addr, (unsigned int)__float_as_int(v));
}

// ---------------- utility fills ----------------
__global__ void k_fill_f32(float* p, float v, long n) {
  long i = (long)blockIdx.x * blockDim.x + threadIdx.x;
  if (i < n) p[i] = v;
}
__global__ void k_fill_u8(unsigned char* p, unsigned char v, long n) {
  long i = (long)blockIdx.x * blockDim.x + threadIdx.x;
  if (i < n) p[i] = v;
}
__global__ void k_set_mask(unsigned char* m, const int* __restrict__ idx, int n) {
  int i = blockIdx.x * blockDim.x + threadIdx.x;
  if (i < n) m[idx[i]] = 1;
}

// ---------------- self-loop attr = mean of incoming edge_attr ----------------
__global__ void k_cnt(const int* __restrict__ EI, float* CNT) {
  int e = blockIdx.x * blockDim.x + threadIdx.x;
  if (e < EE) atomicAdd(CNT + EI[EE + e], 1.0f);
}
__global__ void k_easum(const int* __restrict__ EI, const float* __restrict__ EA, float* SUM) {
  long idx = (long)blockIdx.x * blockDim.x + threadIdx.x;
  if (idx >= (long)EE * FEDGE) return;
  int e = (int)(idx >> 5), k = (int)(idx & 31);
  atomicAdd(SUM + (size_t)EI[EE + e] * FEDGE + k, EA[idx]);
}
__global__ void k_loopdiv(float* SUM, const float* __restrict__ CNT) {
  long idx = (long)blockIdx.x * blockDim.x + threadIdx.x;
  if (idx >= (long)NN * FEDGE) return;
  SUM[idx] = SUM[idx] / fmaxf(CNT[idx >> 5], 1.0f);
}

// ---------------- node transforms: XL = X@Wl + bl, XR = X@Wr + br (WMMA) ------
// grid.x = N/16 row tiles, 128 threads = 4 waves, wave w owns 16-col tile w.
__global__ void k_transform(const float* __restrict__ X, int K,
                            const float* __restrict__ Wl, const float* __restrict__ bl,
                            const float* __restrict__ Wr, const float* __restrict__ br,
                            float* __restrict__ XL, float* __restrict__ XR) {
  const int lane = threadIdx.x & 31;
  const int wave = threadIdx.x >> 5;
  const int l16 = lane & 15, half = lane >> 4;
  const int col0 = wave * 16;
  const int rowBase = blockIdx.x * 16;
  const int arow = rowBase + l16;
  const int col = col0 + l16;

  v8f accL = {0.f,0.f,0.f,0.f,0.f,0.f,0.f,0.f};
  v8f accR = {0.f,0.f,0.f,0.f,0.f,0.f,0.f,0.f};
  for (int k0 = 0; k0 < K; k0 += 4) {
    const int kk = k0 + 2 * half;
    const float2 av = *(const float2*)(X + (size_t)arow * K + kk);
    v2f a; a[0] = av.x; a[1] = av.y;
    v2f bL; bL[0] = Wl[kk * HID + col]; bL[1] = Wl[(kk + 1) * HID + col];
    v2f bR; bR[0] = Wr[kk * HID + col]; bR[1] = Wr[(kk + 1) * HID + col];
    accL = wmma4(a, bL, accL);
    accR = wmma4(a, bR, accR);
  }
  const float vbl = bl[col], vbr = br[col];
#pragma unroll
  for (int r = 0; r < 8; ++r) {
    const int orow = rowBase + r + 8 * half;
    XL[(size_t)orow * HID + col] = accL[r] + vbl;
    XR[(size_t)orow * HID + col] = accR[r] + vbr;
  }
}

// ---------------- fused edge pass A ------------------------------------------
// logits[e] = att . leaky_relu(xl[src]+xr[dst]+ ea_row@We); atomic-max into MX.
// One wave = 16 edges. e = ea@We via 4 col-tiles x 8 K-steps of fp32 WMMA.
__global__ void k_edge_logits(const float* __restrict__ EA, const float* __restrict__ LOOPA,
                              const int* __restrict__ EI,
                              const float* __restrict__ We, const float* __restrict__ att,
                              const float* __restrict__ XL, const float* __restrict__ XR,
                              float* __restrict__ LOG, float* MX,
                              const unsigned char* __restrict__ MASK) {
  const int lane = threadIdx.x & 31, wave = threadIdx.x >> 5;
  const int l16 = lane & 15, half = lane >> 4;
  const long tile = (long)blockIdx.x * (blockDim.x >> 5) + wave;
  if (tile >= NTILES) return;
  const int ebase = (int)tile * 16;

  int sr[8], dr[8];
#pragma unroll
  for (int r = 0; r < 8; ++r) {
    const int e = ebase + r + 8 * half;
    sr[r] = (e < EE) ? EI[e] : (e - EE);
    dr[r] = (e < EE) ? EI[EE + e] : (e - EE);
  }
  if (MASK) {                    // layer-2: skip whole tile if no dst is needed
    bool any = false;
#pragma unroll
    for (int r = 0; r < 8; ++r) any |= (MASK[dr[r]] != 0);
    if (__ballot((int)any) == 0ULL) return;
  }

  // A operand: 16 edge rows x 4 K; preload this lane's 16 floats of ea.
  const int ea_e = ebase + l16;
  const float* ap = (ea_e < EE) ? (EA + (size_t)ea_e * FEDGE)
                                : (LOOPA + (size_t)(ea_e - EE) * FEDGE);
  float2 areg[8];
#pragma unroll
  for (int i = 0; i < 8; ++i) areg[i] = *(const float2*)(ap + i * 4 + 2 * half);

  float s[8] = {0.f,0.f,0.f,0.f,0.f,0.f,0.f,0.f};
#pragma unroll
  for (int c = 0; c < 4; ++c) {
    const int col = c * 16 + l16;
    v8f acc = {0.f,0.f,0.f,0.f,0.f,0.f,0.f,0.f};
#pragma unroll
    for (int i = 0; i < 8; ++i) {
      const int kk = i * 4 + 2 * half;
      v2f a; a[0] = areg[i].x; a[1] = areg[i].y;
      v2f b; b[0] = We[kk * HID + col]; b[1] = We[(kk + 1) * HID + col];
      acc = wmma4(a, b, acc);
    }
    const float ac = att[col];
#pragma unroll
    for (int r = 0; r < 8; ++r) {
      float v = acc[r] + XL[(size_t)sr[r] * HID + col] + XR[(size_t)dr[r] * HID + col];
      v = fmaxf(v, 0.f) + NEG * fminf(v, 0.f);      // leaky_relu
      s[r] += v * ac;
    }
  }
#pragma unroll
  for (int r = 0; r < 8; ++r) {
    float t = s[r];
    t += __shfl_xor(t, 1); t += __shfl_xor(t, 2);
    t += __shfl_xor(t, 4); t += __shfl_xor(t, 8);    // reduce 16 cols -> row
    if (l16 == 0) {
      const int e = ebase + r + 8 * half;
      if (!MASK || MASK[dr[r]]) { LOG[e] = t; atomicMaxF(MX + dr[r], t); }
    }
  }
}

// ---------------- pass B: ex = exp(logit - mx[dst]); den[dst] += ex ----------
__global__ void k_softmax_num(const int* __restrict__ EI, const float* __restrict__ LOG,
                              const float* __restrict__ MX, float* __restrict__ EX,
                              float* DEN, const unsigned char* __restrict__ MASK) {
  const int e = blockIdx.x * blockDim.x + threadIdx.x;
  if (e >= EPRIME) return;
  const int d = (e < EE) ? EI[EE + e] : (e - EE);
  if (MASK && !MASK[d]) return;
  const float ex = expf(LOG[e] - MX[d]);
  EX[e] = ex;
  atomicAdd(DEN + d, ex);
}

// ---------------- pass C: h[dst] += xl[src] * ex/den[dst] --------------------
__global__ void k_aggregate(const int* __restrict__ EI, const float* __restrict__ EX,
                            const float* __restrict__ DEN, const float* __restrict__ XL,
                            float* H, const unsigned char* __restrict__ MASK) {
  const long idx = (long)blockIdx.x * blockDim.x + threadIdx.x;
  if (idx >= (long)EPRIME * 16) return;
  const int e = (int)(idx >> 4), q = (int)(idx & 15);
  const int d = (e < EE) ? EI[EE + e] : (e - EE);
  if (MASK && !MASK[d]) return;
  const int sidx = (e < EE) ? EI[e] : (e - EE);
  const float alpha = EX[e] / DEN[d];
  const float4 xv = *(const float4*)(XL + (size_t)sidx * HID + q * 4);
  float* hp = H + (size_t)d * HID + q * 4;
  atomicAdd(hp + 0, xv.x * alpha);
  atomicAdd(hp + 1, xv.y * alpha);
  atomicAdd(hp + 2, xv.z * alpha);
  atomicAdd(hp + 3, xv.w * alpha);
}

__global__ void k_bias_relu(float* H, const float* __restrict__ bias) {
  const long idx = (long)blockIdx.x * blockDim.x + threadIdx.x;
  if (idx >= (long)NN * HID) return;
  H[idx] = fmaxf(H[idx] + bias[idx & (HID - 1)], 0.f);
}

// ---------------- final MLP on the 1000 gathered rows ------------------------
__global__ void k_mlp(const float* __restrict__ H, const float* __restrict__ Y,
                      const int* __restrict__ IDX,
                      const float* __restrict__ W0, const float* __restrict__ B0,
                      const float* __restrict__ W1, const float* __restrict__ B1,
                      const float* __restrict__ W2, const float* __restrict__ B2,
                      float* __restrict__ OUT) {
  const int lane = threadIdx.x & 31, wave = threadIdx.x >> 5;
  const int row = blockIdx.x * (blockDim.x >> 5) + wave;
  if (row >= NIDX) return;
  const int i = IDX[row];
  const int g = i / REPEAT;
  const float y0 = Y[2 * g], y1 = Y[2 * g + 1];
  const float t0 = fmaxf(y0 * W0[0] + y1 * W0[2] + B0[0], 0.f);
  const float t1 = fmaxf(y0 * W0[1] + y1 * W0[3] + B0[1], 0.f);
  float z1 = B1[lane];
  const float* hp = H + (size_t)i * HID;
  for (int k = 0; k < HID; ++k) z1 += hp[k] * W1[k * ACTD + lane];
  z1 += t0 * W1[HID * ACTD + lane] + t1 * W1[(HID + 1) * ACTD + lane];
  z1 = fmaxf(z1, 0.f);
  float z2 = B2[lane];
  for (int k = 0; k < ACTD; ++k) z2 += __shfl(z1, k) * W2[k * ACTD + lane];
  OUT[row * ACTD + lane] = z2;
}

// ---------------------------------------------------------------------------
extern "C" void kernel_launch(void* const* d_in, const int* in_sizes, int n_in,
                              void* d_out, int out_size, void* d_ws, size_t ws_size,
                              hipStream_t stream) {
  const float* x     = (const float*)d_in[0];
  const float* ea    = (const float*)d_in[1];
  const float* y     = (const float*)d_in[2];
  const float* Wl1   = (const float*)d_in[3];  const float* bl1 = (const float*)d_in[4];
  const float* Wr1   = (const float*)d_in[5];  const float* br1 = (const float*)d_in[6];
  const float* We1   = (const float*)d_in[7];  const float* att1 = (const float*)d_in[8];
  const float* bias1 = (const float*)d_in[9];
  const float* Wl2   = (const float*)d_in[10]; const float* bl2 = (const float*)d_in[11];
  const float* Wr2   = (const float*)d_in[12]; const float* br2 = (const float*)d_in[13];
  const float* We2   = (const float*)d_in[14]; const float* att2 = (const float*)d_in[15];
  const float* bias2 = (const float*)d_in[16];
  const float* W0    = (const float*)d_in[17]; const float* b0 = (const float*)d_in[18];
  const float* W1    = (const float*)d_in[19]; const float* b1 = (const float*)d_in[20];
  const float* W2    = (const float*)d_in[21]; const float* b2 = (const float*)d_in[22];
  const int*   EI    = (const int*)d_in[23];
  const int*   nidx  = (const int*)d_in[24];
  float* out = (float*)d_out;

  char* base = (char*)d_ws;
  size_t off = 0;
  auto alloc = [&](size_t bytes) { size_t o = off; off = (off + bytes + 255) & ~(size_t)255; return o; };
  float* CNT   = (float*)(base + alloc((size_t)NN * 4));
  float* LOOPA = (float*)(base + alloc((size_t)NN * FEDGE * 4));   // sum -> mean in place
  float* XL    = (float*)(base + alloc((size_t)NN * HID * 4));
  float* XR    = (float*)(base + alloc((size_t)NN * HID * 4));
  float* H     = (float*)(base + alloc((size_t)NN * HID * 4));
  float* LOG   = (float*)(base + alloc((size_t)EPRIME * 4));
  float* EX    = (float*)(base + alloc((size_t)EPRIME * 4));
  float* MX    = (float*)(base + alloc((size_t)NN * 4));
  float* DEN   = (float*)(base + alloc((size_t)NN * 4));
  unsigned char* MASK = (unsigned char*)(base + alloc((size_t)NN));

  const int T = 256;
  auto blk  = [](long n, int t) { return (int)((n + t - 1) / t); };
  const float NEGMAX = -3.402823466e38f;

  // self-loop attrs (mean incoming)
  k_fill_f32<<<blk(NN, T), T, 0, stream>>>(CNT, 0.f, NN);
  k_fill_f32<<<blk((long)NN * FEDGE, T), T, 0, stream>>>(LOOPA, 0.f, (long)NN * FEDGE);
  k_cnt<<<blk(EE, T), T, 0, stream>>>(EI, CNT);
  k_easum<<<blk((long)EE * FEDGE, T), T, 0, stream>>>(EI, ea, LOOPA);
  k_loopdiv<<<blk((long)NN * FEDGE, T), T, 0, stream>>>(LOOPA, CNT);

  // dst mask for layer-2 pruning
  k_fill_u8<<<blk(NN, T), T, 0, stream>>>(MASK, 0, NN);
  k_set_mask<<<blk(NIDX, T), T, 0, stream>>>(MASK, nidx, NIDX);

  // ---- layer 1 (all edges) ----
  k_transform<<<NN / 16, 128, 0, stream>>>(x, FNODE, Wl1, bl1, Wr1, br1, XL, XR);
  k_fill_f32<<<blk(NN, T), T, 0, stream>>>(MX, NEGMAX, NN);
  k_fill_f32<<<blk(NN, T), T, 0, stream>>>(DEN, 0.f, NN);
  k_fill_f32<<<blk((long)NN * HID, T), T, 0, stream>>>(H, 0.f, (long)NN * HID);
  k_edge_logits<<<blk(NTILES, 8), 256, 0, stream>>>(ea, LOOPA, EI, We1, att1, XL, XR, LOG, MX, nullptr);
  k_softmax_num<<<blk(EPRIME, T), T, 0, stream>>>(EI, LOG, MX, EX, DEN, nullptr);
  k_aggregate<<<blk((long)EPRIME * 16, T), T, 0, stream>>>(EI, EX, DEN, XL, H, nullptr);
  k_bias_relu<<<blk((long)NN * HID, T), T, 0, stream>>>(H, bias1);

  // ---- layer 2 (masked: only dst in node_idx matter) ----
  k_transform<<<NN / 16, 128, 0, stream>>>(H, HID, Wl2, bl2, Wr2, br2, XL, XR);
  k_fill_f32<<<blk(NN, T), T, 0, stream>>>(MX, NEGMAX, NN);
  k_fill_f32<<<blk(NN, T), T, 0, stream>>>(DEN, 0.f, NN);
  k_fill_f32<<<blk((long)NN * HID, T), T, 0, stream>>>(H, 0.f, (long)NN * HID);
  k_edge_logits<<<blk(NTILES, 8), 256, 0, stream>>>(ea, LOOPA, EI, We2, att2, XL, XR, LOG, MX, MASK);
  k_softmax_num<<<blk(EPRIME, T), T, 0, stream>>>(EI, LOG, MX, EX, DEN, MASK);
  k_aggregate<<<blk((long)EPRIME * 16, T), T, 0, stream>>>(EI, EX, DEN, XL, H, MASK);
  k_bias_relu<<<blk((long)NN * HID, T), T, 0, stream>>>(H, bias2);

  // ---- gathered MLP head ----
  k_mlp<<<blk(NIDX, 8), 256, 0, stream>>>(H, y, nidx, W0, b0, W1, b1, W2, b2, out);
}